// QuanvolutionHybridGraphQL_65481071404068
// MI455X (gfx1250) — compile-verified
//
#include <hip/hip_runtime.h>

// CDNA5 / gfx1250. wave32. One workgroup (256 thr = 8 waves) per image.
// Phase 1: build q (for pooling) and qn (normalized) in LDS.
// Phase 2: tiny MLP head -> logits.
// Phase 3: 196x196 Gram via V_WMMA_F32_16X16X4_F32 (K=4 matches exactly),
//          square+threshold in regs, stream adj to HBM with NT stores.
//          Interior tiles (144) take a branch-free epilogue; only the 25
//          boundary tiles (ti==12 or tj==12) use bounds checks.

typedef __attribute__((ext_vector_type(2))) float v2f;
typedef __attribute__((ext_vector_type(8))) float v8f;

#define NP    196
#define NPAD  208
#define NT    13          // 13*16 = 208 padded tiles per dim
#define NTI   12          // interior tiles per dim
#define THRESH    0.8f
#define SECOND    0.5f
#define SECOND_W  0.5f

__device__ __forceinline__ float adj_val(float g, int row, int col) {
    const float fid = g * g;
    float v = (fid >= THRESH) ? 1.0f : ((fid >= SECOND) ? SECOND_W : 0.0f);
    return (row == col) ? 0.0f : v;   // branchless: v_cmp + v_cndmask
}

__global__ __launch_bounds__(256)
void quanv_fused_kernel(const float* __restrict__ x,    // (B,1,28,28)
                        const float* __restrict__ W1,   // (4,32)
                        const float* __restrict__ b1,   // (32)
                        const float* __restrict__ W2,   // (32,10)
                        const float* __restrict__ b2,   // (10)
                        float* __restrict__ out_logits, // (B,10)
                        float* __restrict__ out_adj)    // (B,196,196)
{
    __shared__ float lds_qn[NPAD][4];   // normalized patches, zero-padded 196..207
    __shared__ float lds_q [NP][4];     // raw cos(patches) for pooling
    __shared__ float lds_pooled[4];
    __shared__ float lds_h[32];

    const int b = blockIdx.x;
    const int t = threadIdx.x;
    const float* img = x + (size_t)b * 784;

    // ---- Phase 1: patches -> q, qn in LDS -------------------------------
    if (t < NPAD) {
        if (t < NP) {
            const int pr = t / 14, pc = t % 14;
            const float* p0 = img + (2 * pr) * 28 + 2 * pc;
            // patch element k = dr*2+dc -> img[2pr+dr][2pc+dc]
            const float q0 = __cosf(p0[0]);
            const float q1 = __cosf(p0[1]);
            const float q2 = __cosf(p0[28]);
            const float q3 = __cosf(p0[29]);
            lds_q[t][0] = q0; lds_q[t][1] = q1; lds_q[t][2] = q2; lds_q[t][3] = q3;
            const float inv = 1.0f / (sqrtf(q0*q0 + q1*q1 + q2*q2 + q3*q3) + 1e-12f);
            lds_qn[t][0] = q0 * inv; lds_qn[t][1] = q1 * inv;
            lds_qn[t][2] = q2 * inv; lds_qn[t][3] = q3 * inv;
        } else {
            lds_qn[t][0] = 0.0f; lds_qn[t][1] = 0.0f;
            lds_qn[t][2] = 0.0f; lds_qn[t][3] = 0.0f;
        }
    }
    __syncthreads();

    // ---- Phase 2: pooled -> 4->32->10 MLP head --------------------------
    if (t < 4) {
        float s = 0.0f;
        for (int p = 0; p < NP; ++p) s += lds_q[p][t];
        lds_pooled[t] = s * (1.0f / 196.0f);
    }
    __syncthreads();
    if (t < 32) {
        float h = b1[t];
        #pragma unroll
        for (int k = 0; k < 4; ++k) h += lds_pooled[k] * W1[k * 32 + t];
        lds_h[t] = h;
    }
    __syncthreads();
    if (t < 10) {
        float lg = b2[t];
        #pragma unroll
        for (int i = 0; i < 32; ++i) lg += lds_h[i] * W2[i * 10 + t];
        out_logits[(size_t)b * 10 + t] = lg;
    }

    // ---- Phase 3: Gram tiles via WMMA f32 16x16x4 -----------------------
    const int wave = t >> 5;
    const int lane = t & 31;
    const int half = lane >> 4;      // 0: lanes 0-15, 1: lanes 16-31
    const int l16  = lane & 15;
    const int kHi  = half * 2;       // K offset {0,2} per lane half
    float* __restrict__ adj = out_adj + (size_t)b * (NP * NP);

    // --- Interior tiles: 144 total, exactly 18 per wave, no bounds checks.
    for (int it = wave; it < NTI * NTI; it += 8) {
        const int ti = it / NTI, tj = it % NTI;
        const int mBase = ti * 16, nBase = tj * 16;

        // A (16x4): lane -> row M=l16, K = {kHi, kHi+1} across 2 VGPRs.
        // B (4x16) = Qn^T: identical per-lane pattern, column base nBase.
        v2f av, bv;
        av.x = lds_qn[mBase + l16][kHi + 0];
        av.y = lds_qn[mBase + l16][kHi + 1];
        bv.x = lds_qn[nBase + l16][kHi + 0];
        bv.y = lds_qn[nBase + l16][kHi + 1];

        v8f c = {};
        c = __builtin_amdgcn_wmma_f32_16x16x4_f32(
                false, av, false, bv, (short)0, c, false, false);

        // D layout: VGPR r -> row (mBase + r + half*8), col (nBase + l16)
        const int col = nBase + l16;
        const int row0 = mBase + half * 8;
        float* __restrict__ p = adj + row0 * NP + col;
        #pragma unroll
        for (int r = 0; r < 8; ++r) {
            __builtin_nontemporal_store(adj_val(c[r], row0 + r, col), p + r * NP);
        }
    }

    // --- Boundary tiles: ti==12 (tj=0..12) then tj==12 (ti=0..11): 25 total.
    for (int et = wave; et < 2 * NT - 1; et += 8) {
        const int ti = (et < NT) ? (NT - 1) : (et - NT);
        const int tj = (et < NT) ? et : (NT - 1);
        const int mBase = ti * 16, nBase = tj * 16;

        v2f av, bv;
        av.x = lds_qn[mBase + l16][kHi + 0];
        av.y = lds_qn[mBase + l16][kHi + 1];
        bv.x = lds_qn[nBase + l16][kHi + 0];
        bv.y = lds_qn[nBase + l16][kHi + 1];

        v8f c = {};
        c = __builtin_amdgcn_wmma_f32_16x16x4_f32(
                false, av, false, bv, (short)0, c, false, false);

        const int col = nBase + l16;
        if (col < NP) {
            #pragma unroll
            for (int r = 0; r < 8; ++r) {
                const int row = mBase + r + half * 8;
                if (row < NP) {
                    __builtin_nontemporal_store(adj_val(c[r], row, col),
                                                adj + row * NP + col);
                }
            }
        }
    }
}

extern "C" void kernel_launch(void* const* d_in, const int* in_sizes, int n_in,
                              void* d_out, int out_size, void* d_ws, size_t ws_size,
                              hipStream_t stream) {
    const float* x  = (const float*)d_in[0];
    const float* W1 = (const float*)d_in[1];
    const float* b1 = (const float*)d_in[2];
    const float* W2 = (const float*)d_in[3];
    const float* b2 = (const float*)d_in[4];

    const int B = in_sizes[0] / (28 * 28);
    float* out_logits = (float*)d_out;                   // (B,10) first in tuple
    float* out_adj    = (float*)d_out + (size_t)B * 10;  // (B,196,196)

    quanv_fused_kernel<<<B, 256, 0, stream>>>(x, W1, b1, W2, b2, out_logits, out_adj);
}